// GATRegressor_27986006901221
// MI455X (gfx1250) — compile-verified
//
#include <hip/hip_runtime.h>
#include <math.h>
#include <stdint.h>

// ---------------------------------------------------------------------------
// GAT regressor for MI455X (gfx1250, wave32).
//  - GEMMs via V_WMMA_F32_16X16X4_F32 (full fp32, matches reference precision)
//  - weight matrix staged per-block into LDS with GLOBAL_LOAD_ASYNC_TO_LDS_B128
//    (ASYNCcnt / s_wait_asynccnt), bank-conflict-free padded stride
//  - segment softmax via fp32 atomic max (int-punning trick) + atomic add
//  - edge aggregation via global_atomic_add_f32 scatter
// ---------------------------------------------------------------------------

#define HIDDEN 128
#define HEADS  4
#define CHAN   32          // HIDDEN / HEADS
#define NEG_SLOPE 0.2f
#define WPAD   136         // padded LDS row stride (floats): rows k,k+2 hit
                           // disjoint 16-bank groups (2*136 % 64 == 16)
#define GEMM_LDS_BYTES (HIDDEN * WPAD * 4)   // 69632 B (dynamic shared)

typedef float v2f __attribute__((ext_vector_type(2)));
typedef float v8f __attribute__((ext_vector_type(8)));

// ---------------------------------------------------------------------------
// C[M x 128] = A[M x 128] * W[128 x 128] (+bias)(+relu)
// grid.x = M/16 (M must be a multiple of 16; 50000 and 512 both are),
// block = 256 threads = 8 waves; wave w computes the 16x16 tile at cols w*16.
// A fragment (16x4 f32): lanes 0-15 -> K={k,k+1}, lanes 16-31 -> K={k+2,k+3},
// row = lane%16.  B fragment mirrored, sourced from LDS.  C: VGPR i ->
// row i+(lane>>4)*8, col lane%16.
// ---------------------------------------------------------------------------
__global__ __launch_bounds__(256) void gat_gemm128_wmma(
    const float* __restrict__ A, const float* __restrict__ W,
    const float* __restrict__ bias, float* __restrict__ C,
    int M, int act)
{
    extern __shared__ float sW[];            // [128][WPAD]
    const int tid = threadIdx.x;

    // ---- async stage of W into LDS: 4096 x b128, 16 per thread -------------
    for (int i = tid; i < (HIDDEN * HIDDEN) / 4; i += 256) {
        const int row = i >> 5;              // 32 float4 chunks per row
        const int c4  = (i & 31) << 2;
        const float* g = W + row * HIDDEN + c4;
        unsigned int l = (unsigned int)(unsigned long long)(uintptr_t)&sW[row * WPAD + c4];
        asm volatile("global_load_async_to_lds_b128 %0, %1, off"
                     :: "v"(l), "v"(g) : "memory");
    }
    asm volatile("s_wait_asynccnt 0" ::: "memory");
    __syncthreads();

    const int lane  = tid & 31;
    const int wave  = tid >> 5;
    const int row0  = blockIdx.x << 4;
    const int col0  = wave << 4;
    const int m     = lane & 15;          // A row / B,C col within tile
    const int khalf = (lane >> 4) << 1;   // 0 or 2

    v8f acc = {};
    const float* __restrict__ Arow = A + (size_t)(row0 + m) * HIDDEN;
    const float* __restrict__ Bcol = sW + col0 + m;

    for (int k = 0; k < HIDDEN; k += 4) {
        v2f a, b;
        a.x = Arow[k + khalf];
        a.y = Arow[k + khalf + 1];
        b.x = Bcol[(k + khalf)     * WPAD];
        b.y = Bcol[(k + khalf + 1) * WPAD];
        acc = __builtin_amdgcn_wmma_f32_16x16x4_f32(
            /*neg_a=*/false, a, /*neg_b=*/false, b,
            /*c_mod=*/(short)0, acc, /*reuse_a=*/false, /*reuse_b=*/false);
    }

    const int rbase = row0 + ((lane >> 4) << 3);
    const int c     = col0 + m;
    const float bv  = bias ? bias[c] : 0.0f;
#pragma unroll
    for (int i = 0; i < 8; ++i) {
        float v = acc[i] + bv;
        if (act) v = fmaxf(v, 0.0f);
        C[(size_t)(rbase + i) * HIDDEN + c] = v;
    }
}

// ---------------------------------------------------------------------------
// Per-(node, head) attention logits: wave32 == one head's 32 channels.
// a_src[n*4+h] = sum_c h[n,h,c]*att_src[h,c]   (same for dst)
// ---------------------------------------------------------------------------
__global__ __launch_bounds__(256) void gat_attn_scores(
    const float* __restrict__ h, const float* __restrict__ att_s,
    const float* __restrict__ att_d, float* __restrict__ a_s,
    float* __restrict__ a_d, int n_nodes)
{
    const int wid  = blockIdx.x * 8 + (threadIdx.x >> 5);
    const int lane = threadIdx.x & 31;
    const int node = wid >> 2;
    const int head = wid & 3;
    if (node >= n_nodes) return;

    const float hv = h[(size_t)node * HIDDEN + head * CHAN + lane];
    float ps = hv * att_s[head * CHAN + lane];
    float pd = hv * att_d[head * CHAN + lane];
#pragma unroll
    for (int off = 16; off > 0; off >>= 1) {
        ps += __shfl_down(ps, off, 32);
        pd += __shfl_down(pd, off, 32);
    }
    if (lane == 0) { a_s[wid] = ps; a_d[wid] = pd; }
}

__global__ void gat_zero_f32(float* __restrict__ p, long long n)
{
    long long i = (long long)blockIdx.x * blockDim.x + threadIdx.x;
    if (i < n) p[i] = 0.0f;
}

__global__ void gat_init_softmax(float* __restrict__ m, float* __restrict__ den, int n)
{
    int i = blockIdx.x * blockDim.x + threadIdx.x;
    if (i < n) { m[i] = -INFINITY; den[i] = 0.0f; }
}

// float atomic max over possibly-negative values via monotone bit patterns
__device__ __forceinline__ void atomic_fmax(float* addr, float v)
{
    if (v >= 0.0f) atomicMax((int*)addr, __float_as_int(v));
    else           atomicMin((unsigned int*)addr, __float_as_uint(v));
}

// pass 1: e = leaky_relu(a_src[src]+a_dst[dst]); segment max into m[dst]
__global__ __launch_bounds__(256) void gat_edge_score_max(
    const float* __restrict__ a_s, const float* __restrict__ a_d,
    const int* __restrict__ ei, float* __restrict__ esc,
    float* __restrict__ mbuf, int nE, int nEtot)
{
    int idx = blockIdx.x * blockDim.x + threadIdx.x;
    if (idx >= nEtot * HEADS) return;
    const int e = idx >> 2, hh = idx & 3;
    int s, d;
    if (e < nE) { s = ei[e]; d = ei[nE + e]; } else { s = d = e - nE; }
    float v = a_s[s * 4 + hh] + a_d[d * 4 + hh];
    v = (v > 0.0f) ? v : v * NEG_SLOPE;
    esc[idx] = v;
    atomic_fmax(&mbuf[d * 4 + hh], v);
}

// pass 2: ex = exp(e - m[dst]); denom[dst] += ex; store ex back over esc
__global__ __launch_bounds__(256) void gat_edge_exp_sum(
    const int* __restrict__ ei, float* __restrict__ esc,
    const float* __restrict__ mbuf, float* __restrict__ den,
    int nE, int nEtot)
{
    int idx = blockIdx.x * blockDim.x + threadIdx.x;
    if (idx >= nEtot * HEADS) return;
    const int e = idx >> 2, hh = idx & 3;
    int d;
    if (e < nE) { d = ei[nE + e]; } else { d = e - nE; }
    float ex = __expf(esc[idx] - mbuf[d * 4 + hh]);
    esc[idx] = ex;
    atomicAdd(&den[d * 4 + hh], ex);
}

// pass 3: out[dst, :] += hT[src, :] * (ex / denom[dst, head])  (2 edges/block)
__global__ __launch_bounds__(256) void gat_edge_aggregate(
    const float* __restrict__ hT, const float* __restrict__ esc,
    const float* __restrict__ den, const int* __restrict__ ei,
    float* __restrict__ out, int nE, int nEtot)
{
    const int e = blockIdx.x * 2 + (threadIdx.x >> 7);
    const int t = threadIdx.x & 127;
    if (e >= nEtot) return;
    int s, d;
    if (e < nE) { s = ei[e]; d = ei[nE + e]; } else { s = d = e - nE; }
    const int head = t >> 5;
    const float alpha = esc[e * 4 + head] / den[d * 4 + head];
    atomicAdd(&out[(size_t)d * HIDDEN + t], hT[(size_t)s * HIDDEN + t] * alpha);
}

__global__ void gat_bias_elu(float* __restrict__ h, const float* __restrict__ b,
                             long long n)
{
    long long i = (long long)blockIdx.x * blockDim.x + threadIdx.x;
    if (i >= n) return;
    float v = h[i] + b[i & (HIDDEN - 1)];
    h[i] = (v > 0.0f) ? v : (__expf(v) - 1.0f);
}

// mean pool accumulation: gsum[batch[n], c] += h[n, c]; gcnt[batch[n]] += 1
__global__ void gat_pool(const float* __restrict__ h, const int* __restrict__ batch,
                         float* __restrict__ gsum, float* __restrict__ gcnt,
                         long long n)
{
    long long i = (long long)blockIdx.x * blockDim.x + threadIdx.x;
    if (i >= n) return;
    const int node = (int)(i >> 7);
    const int c    = (int)(i & 127);
    const int g    = batch[node];
    atomicAdd(&gsum[(size_t)g * HIDDEN + c], h[i]);
    if (c == 0) atomicAdd(&gcnt[g], 1.0f);
}

__global__ void gat_pool_div(float* __restrict__ gsum, const float* __restrict__ gcnt,
                             int n)
{
    int i = blockIdx.x * blockDim.x + threadIdx.x;
    if (i >= n) return;
    gsum[i] /= fmaxf(gcnt[i >> 7], 1.0f);
}

// out[g] = dot(gh[g,:], W2) + b2
__global__ __launch_bounds__(128) void gat_mlp_out(
    const float* __restrict__ gh, const float* __restrict__ W2,
    const float* __restrict__ b2, float* __restrict__ out)
{
    __shared__ float red[4];
    const int g = blockIdx.x, t = threadIdx.x;
    float p = gh[(size_t)g * HIDDEN + t] * W2[t];
#pragma unroll
    for (int off = 16; off > 0; off >>= 1) p += __shfl_down(p, off, 32);
    if ((t & 31) == 0) red[t >> 5] = p;
    __syncthreads();
    if (t == 0) out[g] = red[0] + red[1] + red[2] + red[3] + b2[0];
}

// ---------------------------------------------------------------------------
extern "C" void kernel_launch(void* const* d_in, const int* in_sizes, int n_in,
                              void* d_out, int out_size, void* d_ws, size_t ws_size,
                              hipStream_t stream)
{
    const float* x      = (const float*)d_in[0];
    const int*   ei     = (const int*)  d_in[1];
    const int*   batch  = (const int*)  d_in[3];
    const float* convW[3]  = {(const float*)d_in[4],  (const float*)d_in[8],  (const float*)d_in[12]};
    const float* convAS[3] = {(const float*)d_in[5],  (const float*)d_in[9],  (const float*)d_in[13]};
    const float* convAD[3] = {(const float*)d_in[6],  (const float*)d_in[10], (const float*)d_in[14]};
    const float* convB[3]  = {(const float*)d_in[7],  (const float*)d_in[11], (const float*)d_in[15]};
    const float* mlpW1 = (const float*)d_in[16];
    const float* mlpB1 = (const float*)d_in[17];
    const float* mlpW2 = (const float*)d_in[18];
    const float* mlpB2 = (const float*)d_in[19];
    float* out = (float*)d_out;

    const int nN    = in_sizes[3];          // 50000 nodes (batch vector length)
    const int nE    = in_sizes[1] / 2;      // 800000 edges
    const int nEtot = nE + nN;              // + self loops
    const int nG    = out_size;             // 512 graphs
    const long long NH = (long long)nN * HIDDEN;

    // workspace carve-up (floats)
    float* ws   = (float*)d_ws;
    float* hT   = ws;                       // [N,128] transformed features
    float* hH   = hT  + NH;                 // [N,128] aggregated activations
    float* aS   = hH  + NH;                 // [N,4]
    float* aD   = aS  + (size_t)nN * HEADS;
    float* mBuf = aD  + (size_t)nN * HEADS;
    float* dBuf = mBuf+ (size_t)nN * HEADS;
    float* esc  = dBuf+ (size_t)nN * HEADS; // [Etot,4] scores -> exp scores
    float* gsum = esc + (size_t)nEtot * HEADS;
    float* gcnt = gsum+ (size_t)nG * HIDDEN;
    float* gh   = gcnt+ nG;
    (void)ws_size; (void)n_in;

    const int TB = 256;
    const int nhBlocks   = (int)((NH + TB - 1) / TB);
    const int ehBlocks   = (nEtot * HEADS + TB - 1) / TB;
    const int aggBlocks  = (nEtot + 1) / 2;
    const int attnBlocks = (nN * HEADS + 7) / 8;

    const float* layer_in = x;
    for (int l = 0; l < 3; ++l) {
        // dense transform via WMMA, W staged to LDS with async copies
        gat_gemm128_wmma<<<nN / 16, 256, GEMM_LDS_BYTES, stream>>>(
            layer_in, convW[l], nullptr, hT, nN, 0);
        // per-(node,head) logits
        gat_attn_scores<<<attnBlocks, 256, 0, stream>>>(hT, convAS[l], convAD[l], aS, aD, nN);
        // segment softmax
        gat_init_softmax<<<(nN * HEADS + TB - 1) / TB, TB, 0, stream>>>(mBuf, dBuf, nN * HEADS);
        gat_edge_score_max<<<ehBlocks, TB, 0, stream>>>(aS, aD, ei, esc, mBuf, nE, nEtot);
        gat_edge_exp_sum<<<ehBlocks, TB, 0, stream>>>(ei, esc, mBuf, dBuf, nE, nEtot);
        // aggregate messages
        gat_zero_f32<<<nhBlocks, TB, 0, stream>>>(hH, NH);
        gat_edge_aggregate<<<aggBlocks, 256, 0, stream>>>(hT, esc, dBuf, ei, hH, nE, nEtot);
        // bias + ELU
        gat_bias_elu<<<nhBlocks, TB, 0, stream>>>(hH, convB[l], NH);
        layer_in = hH;
    }

    // global mean pool
    const long long GH = (long long)nG * HIDDEN;
    gat_zero_f32<<<(int)((GH + nG + TB - 1) / TB), TB, 0, stream>>>(gsum, GH + nG); // gsum + gcnt contiguous
    gat_pool<<<nhBlocks, TB, 0, stream>>>(hH, batch, gsum, gcnt, NH);
    gat_pool_div<<<(int)((GH + TB - 1) / TB), TB, 0, stream>>>(gsum, gcnt, (int)GH);

    // MLP head: relu(g @ W1 + b1) @ W2 + b2
    gat_gemm128_wmma<<<nG / 16, 256, GEMM_LDS_BYTES, stream>>>(gsum, mlpW1, mlpB1, gh, nG, 1);
    gat_mlp_out<<<nG, 128, 0, stream>>>(gh, mlpW2, mlpB2, out);
}